// FNN_Model_80668075754315
// MI455X (gfx1250) — compile-verified
//
#include <hip/hip_runtime.h>
#include <math.h>

// Problem constants from the reference
#define B_ 64
#define T_ 20
#define V_ 4130
#define C_ 70

typedef __attribute__((ext_vector_type(2))) float v2f;
typedef __attribute__((ext_vector_type(8))) float v8f;

// ---------------------------------------------------------------------------
// Stage 1: g+[v] = sum_c relu( x[v,c]) * fc0_w[c]
//          g-[v] = sum_c relu(-x[v,c]) * fc0_w[c]
// One wave32 per v. 1.2 MB read, trivial.
// ---------------------------------------------------------------------------
__global__ void precompute_g_kernel(const float* __restrict__ embed_table,
                                    const float* __restrict__ fc0_w,
                                    float* __restrict__ g_pos,
                                    float* __restrict__ g_neg) {
    const int lane = threadIdx.x & 31;
    const int v    = blockIdx.x * (blockDim.x >> 5) + (threadIdx.x >> 5);
    if (v >= V_) return;  // uniform per-wave exit

    float ap = 0.0f, an = 0.0f;
    for (int c = lane; c < C_; c += 32) {
        float x = embed_table[v * C_ + c];
        float w = fc0_w[c];
        ap = fmaf(fmaxf( x, 0.0f), w, ap);
        an = fmaf(fmaxf(-x, 0.0f), w, an);
    }
    // wave32 butterfly reduce
    #pragma unroll
    for (int off = 16; off > 0; off >>= 1) {
        ap += __shfl_xor(ap, off, 32);
        an += __shfl_xor(an, off, 32);
    }
    if (lane == 0) {
        g_pos[v] = ap;
        g_neg[v] = an;
    }
}

// ---------------------------------------------------------------------------
// Stage 2: out1[row] = relu( sum_v relu(|s|*g±[v] + fc0_b) * fc1_w[v] + fc1_b )
// row = b*T + t; streams visits (21 MB) => the memory-bound stage.
// 1280 blocks x 256 threads (8 wave32s each). Straight-line inner loop:
// both g arrays loaded unconditionally (L2-resident), sign pick via cndmask.
// ---------------------------------------------------------------------------
__global__ void row_reduce_kernel(const float* __restrict__ visits,
                                  const float* __restrict__ g_pos,
                                  const float* __restrict__ g_neg,
                                  const float* __restrict__ fc1_w,
                                  const float* __restrict__ fc0_b,
                                  const float* __restrict__ fc1_b,
                                  float* __restrict__ out1) {
    __shared__ float red[8];
    const int   row = blockIdx.x;
    const float b0  = fc0_b[0];
    const float* __restrict__ vrow = visits + (size_t)row * V_;

    float acc = 0.0f;
    for (int v = threadIdx.x; v < V_; v += 256) {
        float s  = vrow[v];
        float gp = g_pos[v];
        float gn = g_neg[v];
        float gv = (s >= 0.0f) ? gp : gn;   // exact relu(x*s) factorization
        float e  = fmaxf(fmaf(fabsf(s), gv, b0), 0.0f);
        acc = fmaf(e, fc1_w[v], acc);
    }
    // wave32 reduce
    #pragma unroll
    for (int off = 16; off > 0; off >>= 1)
        acc += __shfl_xor(acc, off, 32);
    const int lane = threadIdx.x & 31;
    const int wav  = threadIdx.x >> 5;
    if (lane == 0) red[wav] = acc;
    __syncthreads();
    if (threadIdx.x == 0) {
        float a = 0.0f;
        #pragma unroll
        for (int w = 0; w < 8; ++w) a += red[w];
        out1[row] = fmaxf(a + fc1_b[0], 0.0f);
    }
}

// ---------------------------------------------------------------------------
// Stage 3: emb = relu(out1(64x20) @ fc2_w^T(20x8) + fc2_b)  via WMMA f32 16x16x4
//          logits[b] = sigmoid(emb[b,:] . mlp_w + mlp_b)
// One block, 4 wave32s; wave w owns M-tile rows [16w, 16w+16).
// A 16x4 layout (ISA 7.12.2): lanes 0-15 hold K={0,1}, lanes 16-31 hold K={2,3}.
// C/D 16x16 layout: VGPR j -> row j (lanes 0-15) / row j+8 (lanes 16-31), N=lane&15.
// B loads use clamped index (lm&7) + value select => branchless, EXEC all-ones.
// ---------------------------------------------------------------------------
__global__ void head_kernel(const float* __restrict__ out1,
                            const float* __restrict__ fc2_w,
                            const float* __restrict__ fc2_b,
                            const float* __restrict__ mlp_w,
                            const float* __restrict__ mlp_b,
                            float* __restrict__ d_out) {
    __shared__ float emb_s[B_ * 8];

    const int  lane   = threadIdx.x & 31;
    const int  wav    = threadIdx.x >> 5;   // 0..3
    const int  lm     = lane & 15;          // N column / M row selector
    const int  hi     = lane >> 4;          // lane-half: K offset +2, M offset +8
    const int  mbase  = wav * 16;
    const bool active = (lm < 8);           // columns >= 8 are dead
    const int  ncol   = lm & 7;             // clamped, always-in-bounds column

    // C initialized with the broadcast column bias
    v8f c;
    const float cinit = active ? fc2_b[ncol] : 0.0f;
    #pragma unroll
    for (int j = 0; j < 8; ++j) c[j] = cinit;

    // K = 20 as 5 chunks of 4; unconditional loads + value selects only
    #pragma unroll
    for (int kb = 0; kb < T_; kb += 4) {
        const int k0 = kb + 2 * hi;
        v2f a, b;
        a.x = out1[(mbase + lm) * T_ + k0];
        a.y = out1[(mbase + lm) * T_ + k0 + 1];
        // B[k][n] = fc2_w^T[k][n] = fc2_w[n*T + k]; zero for n >= 8 via cndmask
        float w0 = fc2_w[ncol * T_ + k0];
        float w1 = fc2_w[ncol * T_ + k0 + 1];
        b.x = active ? w0 : 0.0f;
        b.y = active ? w1 : 0.0f;
        c = __builtin_amdgcn_wmma_f32_16x16x4_f32(
                /*neg_a=*/false, a, /*neg_b=*/false, b,
                /*c_mod=*/(short)0, c, /*reuse_a=*/false, /*reuse_b=*/false);
    }

    // relu + scatter D fragment: VGPR j holds row (mbase + j + 8*hi), col lm
    #pragma unroll
    for (int j = 0; j < 8; ++j) {
        const int m = mbase + j + 8 * hi;
        const float e = fmaxf(c[j], 0.0f);
        if (active) {
            emb_s[m * 8 + lm]      = e;
            d_out[B_ + m * 8 + lm] = e;   // emb output (after 64 logits)
        }
    }
    __syncthreads();

    // sigmoid head: one thread per batch row
    if (threadIdx.x < B_) {
        float z = mlp_b[0];
        #pragma unroll
        for (int j = 0; j < 8; ++j)
            z = fmaf(emb_s[threadIdx.x * 8 + j], mlp_w[j], z);
        d_out[threadIdx.x] = 1.0f / (1.0f + expf(-z));   // logits output
    }
}

// ---------------------------------------------------------------------------
extern "C" void kernel_launch(void* const* d_in, const int* in_sizes, int n_in,
                              void* d_out, int out_size, void* d_ws, size_t ws_size,
                              hipStream_t stream) {
    const float* visits      = (const float*)d_in[0];
    const float* embed_table = (const float*)d_in[1];
    const float* fc0_w       = (const float*)d_in[2];
    const float* fc0_b       = (const float*)d_in[3];
    const float* fc1_w       = (const float*)d_in[4];
    const float* fc1_b       = (const float*)d_in[5];
    const float* fc2_w       = (const float*)d_in[6];
    const float* fc2_b       = (const float*)d_in[7];
    const float* mlp_w       = (const float*)d_in[8];
    const float* mlp_b       = (const float*)d_in[9];
    float* out = (float*)d_out;

    // workspace layout (16B-aligned slots): g_pos[4160] | g_neg[4160] | out1[1280]
    float* g_pos = (float*)d_ws;
    float* g_neg = g_pos + 4160;
    float* out1  = g_neg + 4160;

    // Stage 1: one wave per v, 8 waves per block
    precompute_g_kernel<<<(V_ + 7) / 8, 256, 0, stream>>>(embed_table, fc0_w, g_pos, g_neg);
    // Stage 2: one block per (b,t) row — streams visits at HBM rate
    row_reduce_kernel<<<B_ * T_, 256, 0, stream>>>(visits, g_pos, g_neg, fc1_w,
                                                   fc0_b, fc1_b, out1);
    // Stage 3: single block, 4 waves, WMMA f32 GEMM + sigmoid head
    head_kernel<<<1, 128, 0, stream>>>(out1, fc2_w, fc2_b, mlp_w, mlp_b, out);
}